// SmallLesionFocalLoss_45646912422162
// MI455X (gfx1250) — compile-verified
//
#include <hip/hip_runtime.h>
#include <hip/hip_bf16.h>

// Problem constants (from reference setup_inputs / hyper-params)
#define BATCH 4
#define DD 128
#define HHH 160
#define WWW 160
#define HW (HHH * WWW)                 // 25600
#define NVOL (DD * HW)                 // 3,276,800 (divisible by 4)
#define NTOT (BATCH * NVOL)            // 13,107,200
#define THR_SIZE 50
#define W_SMALL 3.0f
#define ALPHA_F 0.25f
#define DICE_W 0.6f
#define FOCAL_W 0.4f
#define SMOOTH 1e-5f
#define PROP_ITERS 10
#define TPB 256
#define VPT 4
#define TILE (TPB * VPT)               // 1024 elements = 4 KB per buffer
#define NBLK_TILE (NTOT / TILE)        // 12,800 blocks, no tail -> EXEC all ones
#define NBLK_ALL (NTOT / TPB)          // 51,200 blocks (scalar kernels)
#define NBLK_V4 (NTOT / 4 / TPB)       // 12,800 blocks (b128 kernels)

typedef __attribute__((ext_vector_type(2))) float v2f;
typedef __attribute__((ext_vector_type(8))) float v8f;
typedef __attribute__((ext_vector_type(4))) unsigned int v4u;
typedef __attribute__((ext_vector_type(4))) int v4i;
typedef __attribute__((ext_vector_type(8))) int v8i;

__device__ __forceinline__ int imin(int a, int b) { return a < b ? a : b; }

// ---------------------------------------------------------------------------
// Tensor Data Mover: 1-D tile load, global -> LDS. D# per CDNA5 ISA §8:
//  group0: [1:0]=count=1 | lds_addr@[63:32] | global_addr@[120:64] | type=2@[127:126]
//  group1: data_size=4B@[17:16] | tensor_dim0@[79:48] | tensor_dim1=1@[111:80]
//          | tile_dim0@[127:112] | tensor_dim0_stride@[207:160]
//  groups 2/3: zero (<=2D tensor). Completion tracked with TENSORcnt.
// ---------------------------------------------------------------------------
__device__ __forceinline__ void tdm_load_1d(const void* gptr, unsigned lds_off,
                                            unsigned nelem) {
    unsigned long long ga = (unsigned long long)(uintptr_t)gptr; // 57-bit VA
    v4u g0;
    g0[0] = 1u;                                       // count=1 (valid), user mode
    g0[1] = lds_off;                                  // lds_addr (bytes)
    g0[2] = (unsigned)(ga & 0xffffffffu);             // global_addr[31:0]
    g0[3] = (unsigned)((ga >> 32) & 0x01ffffffu)      // global_addr[56:32]
          | 0x80000000u;                              // type=2 ("image")
    v8i g1;
    g1[0] = (int)(2u << 16);                          // data_size=2 -> 4 bytes
    g1[1] = (int)((nelem & 0xffffu) << 16);           // tensor_dim0[15:0]
    g1[2] = (int)((nelem >> 16) | (1u << 16));        // tensor_dim0[31:16] | tensor_dim1=1
    g1[3] = (int)((nelem & 0xffffu) << 16);           // tile_dim0
    g1[4] = 0;                                        // tile_dim1=0, tile_dim2=0 (unused)
    g1[5] = (int)nelem;                               // tensor_dim0_stride[31:0]
    g1[6] = 0;                                        // stride hi | dim1_stride lo
    g1[7] = 0;
    v4i z4 = {0, 0, 0, 0};
#if defined(__clang_major__) && (__clang_major__ >= 23)
    v8i z8 = {0, 0, 0, 0, 0, 0, 0, 0};
    __builtin_amdgcn_tensor_load_to_lds(g0, g1, z4, z4, z8, 0);
#else
    __builtin_amdgcn_tensor_load_to_lds(g0, g1, z4, z4, 0);
#endif
}

// Low 32 bits of a generic pointer into the LDS aperture == LDS byte offset.
__device__ __forceinline__ unsigned lds_off_of(const void* p) {
    return (unsigned)(uintptr_t)p;
}

// ---------------------------------------------------------------------------
// Full-wave (32 lane) f32 sum on the matrix pipe (V_WMMA_F32_16X16X4_F32).
// A=(v,0) per lane, B=ones -> D[m][n] = v_m + v_{m+16}; sum 8 C regs per lane
// (rows m..m+7), then xor-16 to combine halves. Requires EXEC all ones.
// ---------------------------------------------------------------------------
__device__ __forceinline__ float wave_sum_wmma(float v) {
    v2f A; A[0] = v;    A[1] = 0.0f;
    v2f B; B[0] = 1.0f; B[1] = 1.0f;
    v8f C = {};
    C = __builtin_amdgcn_wmma_f32_16x16x4_f32(false, A, false, B,
                                              (short)0, C, false, false);
    float s = C[0] + C[1] + C[2] + C[3] + C[4] + C[5] + C[6] + C[7];
    s += __shfl_xor(s, 16, 32);
    return s;
}

// ---------------------------------------------------------------------------
// K0: zero counts (b128 stores) + 4 global accumulators
// ---------------------------------------------------------------------------
__global__ void k_zero(int4* __restrict__ counts4, float* __restrict__ accum) {
    int q = blockIdx.x * blockDim.x + threadIdx.x;      // NTOT/4 threads
    int4 z; z.x = 0; z.y = 0; z.z = 0; z.w = 0;
    counts4[q] = z;
    if (q == 0) { accum[0] = 0.f; accum[1] = 0.f; accum[2] = 0.f; accum[3] = 0.f; }
}

// ---------------------------------------------------------------------------
// K1: init labels (b128 load/store): fg -> local flat index, bg -> NVOL
// ---------------------------------------------------------------------------
__global__ void k_init(const float4* __restrict__ tgt4, int4* __restrict__ label4) {
    int q = blockIdx.x * blockDim.x + threadIdx.x;
    float4 t = tgt4[q];
    int i = q * 4;                   // 4 consecutive elems, same volume (NVOL%4==0)
    int vol = i / NVOL;
    int loc = i - vol * NVOL;
    int4 o;
    o.x = (t.x > 0.5f) ? loc     : NVOL;
    o.y = (t.y > 0.5f) ? loc + 1 : NVOL;
    o.z = (t.z > 0.5f) ? loc + 2 : NVOL;
    o.w = (t.w > 0.5f) ? loc + 3 : NVOL;
    label4[q] = o;
}

// ---------------------------------------------------------------------------
// K2: one CCL sweep: 6-neighbor min (volume-local), root chase, atomicMin union
// ---------------------------------------------------------------------------
__global__ void k_propagate(int* __restrict__ label) {
    int i = blockIdx.x * blockDim.x + threadIdx.x;
    int l = label[i];
    if (l >= NVOL) return;                 // background: one load, exit
    int vol  = i / NVOL;
    int base = vol * NVOL;
    int loc  = i - base;
    int z = loc / HW;
    int rem = loc - z * HW;
    int y = rem / WWW;
    int x = rem - y * WWW;

    int m = l;
    if (x > 0)       m = imin(m, label[i - 1]);
    if (x < WWW - 1) m = imin(m, label[i + 1]);
    if (y > 0)       m = imin(m, label[i - WWW]);
    if (y < HHH - 1) m = imin(m, label[i + WWW]);
    if (z > 0)       m = imin(m, label[i - HW]);
    if (z < DD - 1)  m = imin(m, label[i + HW]);

    int r = m, p;
    while ((p = label[base + r]) < r) r = p;      // root of neighborhood min
    int rl = l;
    while ((p = label[base + rl]) < rl) rl = p;   // own root
    if (r < rl) atomicMin(&label[base + rl], r);  // union at the roots
    if (r < l)  atomicMin(&label[i], r);
}

// ---------------------------------------------------------------------------
// K3: flatten to root and count component sizes
// ---------------------------------------------------------------------------
__global__ void k_count(int* __restrict__ label, int* __restrict__ counts) {
    int i = blockIdx.x * blockDim.x + threadIdx.x;
    int l = label[i];
    if (l >= NVOL) return;
    int vol  = i / NVOL;
    int base = vol * NVOL;
    int r = l, p;
    while ((p = label[base + r]) < r) r = p;
    label[i] = r;
    atomicAdd(&counts[base + r], 1);
}

// ---------------------------------------------------------------------------
// K4: TDM-staged fused dice+focal. Wave 0 issues two TENSOR_LOAD_TO_LDS (4 KB
//     tiles of pred/target), waits TENSORcnt, workgroup barrier, then all 8
//     waves compute 4 elems/thread from LDS. WMMA wave reduce + LDS combine.
//     accum[0]=sum(p*t*w) accum[1]=sum(p*w) accum[2]=sum(t*w) accum[3]=focal
// ---------------------------------------------------------------------------
__global__ void __launch_bounds__(TPB)
k_loss(const float* __restrict__ pred, const float* __restrict__ tgt,
       const int* __restrict__ label, const int* __restrict__ counts,
       float* __restrict__ accum) {
    __shared__ float s_p[TILE];
    __shared__ float s_t[TILE];
    int tile0 = blockIdx.x * TILE;

    if (threadIdx.x < 32) {                         // wave 0 drives the TDM
        tdm_load_1d(pred + tile0, lds_off_of(s_p), TILE);
        tdm_load_1d(tgt  + tile0, lds_off_of(s_t), TILE);
        __builtin_amdgcn_s_wait_tensorcnt(0);
    }
    __syncthreads();

    float s0 = 0.f, s1 = 0.f, s2 = 0.f, s3 = 0.f;
    #pragma unroll
    for (int k = 0; k < VPT; ++k) {
        int j = k * TPB + threadIdx.x;              // stride-256: conflict-free
        int i = tile0 + j;
        float pv = s_p[j];
        float tv = s_t[j];

        float w = 1.0f;
        if (tv > 0.5f) {
            int vol  = i / NVOL;
            int base = vol * NVOL;
            int c = counts[base + label[i]];
            w = (c < THR_SIZE) ? W_SMALL : 1.0f;
        }

        s0 += pv * tv * w;
        s1 += pv * w;
        s2 += tv * w;

        float sp = fmaxf(-pv, 0.0f) + log1pf(__expf(-fabsf(pv)));
        float ce = (1.0f - tv) * pv + sp;
        float pt = __expf(-ce);
        float om = 1.0f - pt;
        float at = ALPHA_F * tv + (1.0f - ALPHA_F) * (1.0f - tv);
        s3 += at * om * om * ce * w;
    }

    float r0 = wave_sum_wmma(s0);
    float r1 = wave_sum_wmma(s1);
    float r2 = wave_sum_wmma(s2);
    float r3 = wave_sum_wmma(s3);

    __shared__ float smem[TPB / 32][4];
    int wave = threadIdx.x >> 5;
    int lane = threadIdx.x & 31;
    if (lane == 0) {
        smem[wave][0] = r0; smem[wave][1] = r1;
        smem[wave][2] = r2; smem[wave][3] = r3;
    }
    __syncthreads();
    if (threadIdx.x == 0) {
        float t0 = 0.f, t1 = 0.f, t2 = 0.f, t3 = 0.f;
        #pragma unroll
        for (int k = 0; k < TPB / 32; ++k) {
            t0 += smem[k][0]; t1 += smem[k][1];
            t2 += smem[k][2]; t3 += smem[k][3];
        }
        atomicAdd(&accum[0], t0);
        atomicAdd(&accum[1], t1);
        atomicAdd(&accum[2], t2);
        atomicAdd(&accum[3], t3);
    }
}

// ---------------------------------------------------------------------------
// K5: combine to scalar loss
// ---------------------------------------------------------------------------
__global__ void k_final(const float* __restrict__ accum, float* __restrict__ out) {
    if (blockIdx.x == 0 && threadIdx.x == 0) {
        float inter = accum[0];
        float denom = accum[1] + accum[2];
        float dice  = 1.0f - (2.0f * inter + SMOOTH) / (denom + SMOOTH);
        float focal = accum[3] / (float)NTOT;
        out[0] = DICE_W * dice + FOCAL_W * focal;
    }
}

extern "C" void kernel_launch(void* const* d_in, const int* in_sizes, int n_in,
                              void* d_out, int out_size, void* d_ws, size_t ws_size,
                              hipStream_t stream) {
    const float* pred = (const float*)d_in[0];
    const float* tgt  = (const float*)d_in[1];

    int*   label  = (int*)d_ws;              // 52.4 MB
    int*   counts = label + NTOT;            // 52.4 MB
    float* accum  = (float*)(counts + NTOT); // 16 B
    float* out    = (float*)d_out;

    dim3 blk(TPB);

    k_zero<<<dim3(NBLK_V4), blk, 0, stream>>>((int4*)counts, accum);
    k_init<<<dim3(NBLK_V4), blk, 0, stream>>>((const float4*)tgt, (int4*)label);
    for (int it = 0; it < PROP_ITERS; ++it)
        k_propagate<<<dim3(NBLK_ALL), blk, 0, stream>>>(label);
    k_count<<<dim3(NBLK_ALL), blk, 0, stream>>>(label, counts);
    k_loss<<<dim3(NBLK_TILE), blk, 0, stream>>>(pred, tgt, label, counts, accum);
    k_final<<<1, 1, 0, stream>>>(accum, out);
}